// Mod_Net_2662879723598
// MI455X (gfx1250) — compile-verified
//
#include <hip/hip_runtime.h>
#include <cstdint>

#define NUM_CLASSES 101
#define OUT_DIM 128
#define TABLE_ELEMS (NUM_CLASSES * OUT_DIM)   // 12928 floats = 51712 B LDS

// out[b, j] = sigmoid(W[j, x[b]]),  W row-major [128, 101]
//
// Only 101 distinct output rows exist. Build sigmoid(W^T) as a 101x128 f32
// table in LDS once per workgroup, then stream rows to HBM with the gfx1250
// async data mover (global_store_async_from_lds_b128): one instruction per
// wave moves one 512 B output row (32 lanes x 16 B) LDS -> HBM with no VGPR
// round-trip. 512 MB of stores @ 23.3 TB/s ~ 22 us is the only limiter.
__global__ __launch_bounds__(256) void onehot_sigmoid_rows(
    const int* __restrict__ x,
    const float* __restrict__ W,
    float* __restrict__ out,
    int batch)
{
    __shared__ float table[TABLE_ELEMS];

    const int tid = threadIdx.x;

    // ---- Phase 1: sigmoid table (per-workgroup, trivial cost) ----
    for (int t = tid; t < TABLE_ELEMS; t += blockDim.x) {
        const int c = t >> 7;      // class   0..100
        const int j = t & 127;     // feature 0..127
        const float w = W[j * NUM_CLASSES + c];
        const float e = __expf(-w);
        table[t] = __builtin_amdgcn_rcpf(1.0f + e);   // fast v_rcp_f32
    }
    __syncthreads();

    // Wave-relative LDS byte address of the table: low 32 bits of the
    // generic pointer (LDS aperture maps addr[31:0] directly).
    const unsigned lds_base = (unsigned)(uintptr_t)(void*)&table[0];

    // ---- Phase 2: 32 consecutive rows per wave per outer iteration ----
    const int lane          = tid & 31;
    const int waveInBlock   = tid >> 5;
    const int wavesPerBlock = blockDim.x >> 5;
    const int gwave         = blockIdx.x * wavesPerBlock + waveInBlock;
    const int nwaves        = gridDim.x * wavesPerBlock;
    const unsigned laneByte = (unsigned)lane * 16u;
    const int stride32      = nwaves << 5;

    for (int b0 = gwave << 5; b0 + 32 <= batch; b0 += stride32) {
        // One coalesced 128 B load: 32 class ids for 32 consecutive rows.
        const int clsv = x[b0 + lane];

        // 32 back-to-back async row stores, no memory waits in between.
        // Wave writes 16 KB of contiguous HBM per outer iteration.
#pragma unroll
        for (int i = 0; i < 32; ++i) {
            const int cls = __builtin_amdgcn_readlane(clsv, i);   // -> SGPR
            const unsigned voff   = (unsigned)(b0 + i) * 512u + laneByte;
            const unsigned ldsoff = lds_base + (unsigned)cls * 512u + laneByte;
            // GVS addressing: mem = SGPR64(out) + VGPR_I32(voff); NT: the
            // 512 MB stream exceeds 192 MB L2, don't pollute it.
            asm volatile("global_store_async_from_lds_b128 %0, %1, %2 th:TH_STORE_NT"
                         :
                         : "v"(voff), "v"(ldsoff), "s"(out)
                         : "memory");
        }
    }

    // Tail rows (batch not a multiple of 32): one row per leading wave.
    const int btail = (batch & ~31) + gwave;
    if (btail < batch) {
        const int cls = x[btail];
        const unsigned voff   = (unsigned)btail * 512u + laneByte;
        const unsigned ldsoff = lds_base + (unsigned)cls * 512u + laneByte;
        asm volatile("global_store_async_from_lds_b128 %0, %1, %2 th:TH_STORE_NT"
                     :
                     : "v"(voff), "v"(ldsoff), "s"(out)
                     : "memory");
    }

    // Drain outstanding async stores (s_endpgm also implies wait-idle).
    asm volatile("s_wait_asynccnt 0x0" ::: "memory");
}

extern "C" void kernel_launch(void* const* d_in, const int* in_sizes, int n_in,
                              void* d_out, int out_size, void* d_ws, size_t ws_size,
                              hipStream_t stream) {
    (void)n_in; (void)out_size; (void)d_ws; (void)ws_size;

    const int*   x   = (const int*)d_in[0];    // [BATCH] int32 class ids
    const float* W   = (const float*)d_in[1];  // [128, 101] f32 row-major
    float*       out = (float*)d_out;          // [BATCH, 128] f32
    const int batch  = in_sizes[0];

    // 2048 blocks x 8 waves = 16384 waves; 1M rows -> 2 outer iterations
    // of 32 rows (16 KB contiguous) per wave.
    const int threads = 256;
    const int blocks  = 2048;
    onehot_sigmoid_rows<<<blocks, threads, 0, stream>>>(x, W, out, batch);
}